// RandomForest_14637248544926
// MI455X (gfx1250) — compile-verified
//
#include <hip/hip_runtime.h>

// ---------------------------------------------------------------------------
// mean_t(x @ W[t] + b[t])  ==  x @ mean_t(W[t]) + mean_t(b[t])
// Kernel 1: reduce W[64,128,32] -> Wavg[128,32], b[64,32] -> bavg[32] (in d_ws)
// Kernel 2: GEMM [65536,128] x [128,32] via V_WMMA_F32_16X16X4_F32 (fp32 exact)
// ---------------------------------------------------------------------------

typedef __attribute__((ext_vector_type(2))) float v2f;
typedef __attribute__((ext_vector_type(4))) float v4f;
typedef __attribute__((ext_vector_type(8))) float v8f;

#define N_TREES   64
#define N_FEAT    128
#define N_CLS     32
#define BATCH     65536

#define WAVES_PER_BLOCK 4
#define ROWS_PER_WAVE   16
#define THREADS_PER_BLOCK (WAVES_PER_BLOCK * 32)
#define ROWS_PER_BLOCK  (WAVES_PER_BLOCK * ROWS_PER_WAVE)

// Padded LDS row strides (in floats). 132 floats = 528 B:
//  - multiple of 16 B (float4 staging stores stay aligned)
//  - fragment reads at word stride 132 land on 4c+k banks -> 32 distinct banks
#define LDS_STRIDE 132

// ---------------------------------------------------------------------------
// Kernel 1: average weights / biases over trees.
// ---------------------------------------------------------------------------
__global__ void forest_avg_wb(const float* __restrict__ W,
                              const float* __restrict__ b,
                              float* __restrict__ Wavg,
                              float* __restrict__ bavg) {
    const int idx = blockIdx.x * blockDim.x + threadIdx.x;
    if (idx < N_FEAT * N_CLS) {
        float s = 0.0f;
        #pragma unroll 8
        for (int t = 0; t < N_TREES; ++t)
            s += W[t * (N_FEAT * N_CLS) + idx];   // coalesced across idx
        Wavg[idx] = s * (1.0f / N_TREES);
    }
    if (idx < N_CLS) {
        float s = 0.0f;
        #pragma unroll 8
        for (int t = 0; t < N_TREES; ++t)
            s += b[t * N_CLS + idx];
        bavg[idx] = s * (1.0f / N_TREES);
    }
}

// ---------------------------------------------------------------------------
// Kernel 2: X[65536,128] @ Wavg[128,32] + bavg, fp32 WMMA.
// Each wave owns a 16-row tile and both 16-col halves of the output.
// ---------------------------------------------------------------------------
__global__ void __launch_bounds__(THREADS_PER_BLOCK)
forest_gemm(const float* __restrict__ X,
            const float* __restrict__ Wavg,
            const float* __restrict__ bavg,
            float* __restrict__ Out) {
    // B transposed: bt[c][k], c in [0,32), k in [0,128), padded stride.
    __shared__ __align__(16) float lds_bt[N_CLS * LDS_STRIDE];
    // A tiles: one 16x128 tile per wave, padded stride.
    __shared__ __align__(16) float lds_a[WAVES_PER_BLOCK * ROWS_PER_WAVE * LDS_STRIDE];

    const int tid  = threadIdx.x;
    const int wave = tid >> 5;      // wave32
    const int lane = tid & 31;

    // ---- Stage Wavg transposed into LDS (coalesced global reads) ----------
    for (int i = tid; i < N_FEAT * N_CLS; i += THREADS_PER_BLOCK) {
        const int k = i >> 5;       // feature row
        const int c = i & 31;       // class col
        lds_bt[c * LDS_STRIDE + k] = Wavg[i];
    }

    // ---- Stage this wave's 16x128 A tile (coalesced float4 reads) ---------
    const int row_base = (blockIdx.x * WAVES_PER_BLOCK + wave) * ROWS_PER_WAVE;
    const float* a_src = X + (size_t)row_base * N_FEAT;
    float* a_dst = lds_a + wave * ROWS_PER_WAVE * LDS_STRIDE;
    #pragma unroll
    for (int i = lane; i < ROWS_PER_WAVE * (N_FEAT / 4); i += 32) {
        const int r  = i >> 5;      // row within tile (32 float4 per row)
        const int c4 = i & 31;      // float4 index within row
        const v4f v = *(const v4f*)(a_src + r * N_FEAT + c4 * 4);
        *(v4f*)(a_dst + r * LDS_STRIDE + c4 * 4) = v;
    }

    __syncthreads();

    // ---- Fragment addressing per CDNA5 32-bit WMMA layouts -----------------
    // A 16x4 (MxK): lane l holds A[M=l&15][K=klo], A[M=l&15][K=klo+1],
    //               klo = (l>>4)*2  -> contiguous v2f in row-major tile.
    // B 4x16 (KxN): lane l holds B[K=klo][N=l&15], B[K=klo+1][N=l&15]
    //               -> contiguous v2f in transposed bt[c][k] storage.
    const int m   = lane & 15;
    const int klo = (lane >> 4) << 1;

    const float* a_frag  = a_dst  + m * LDS_STRIDE + klo;
    const float* b0_frag = lds_bt + m * LDS_STRIDE + klo;          // cols 0..15
    const float* b1_frag = lds_bt + (m + 16) * LDS_STRIDE + klo;   // cols 16..31

    v8f acc0 = {};   // output cols 0..15
    v8f acc1 = {};   // output cols 16..31

    #pragma unroll
    for (int k = 0; k < N_FEAT; k += 4) {
        const v2f a  = *(const v2f*)(a_frag  + k);
        const v2f b0 = *(const v2f*)(b0_frag + k);
        const v2f b1 = *(const v2f*)(b1_frag + k);
        acc0 = __builtin_amdgcn_wmma_f32_16x16x4_f32(
            /*neg_a=*/false, a, /*neg_b=*/false, b0,
            /*c_mod=*/(short)0, acc0, /*reuse_a=*/false, /*reuse_b=*/false);
        acc1 = __builtin_amdgcn_wmma_f32_16x16x4_f32(
            /*neg_a=*/false, a, /*neg_b=*/false, b1,
            /*c_mod=*/(short)0, acc1, /*reuse_a=*/false, /*reuse_b=*/false);
    }

    // ---- Epilogue: bias add + store -----------------------------------
    // C/D 16x16 f32 layout: VGPR v -> M = v + (lane>=16 ? 8 : 0), N = lane&15.
    const float bias0 = bavg[m];
    const float bias1 = bavg[m + 16];
    const int m_off = (lane >> 4) << 3;
    float* out_base = Out + (size_t)(row_base + m_off) * N_CLS + m;
    #pragma unroll
    for (int i = 0; i < 8; ++i) {
        out_base[(size_t)i * N_CLS]      = acc0[i] + bias0;
        out_base[(size_t)i * N_CLS + 16] = acc1[i] + bias1;
    }
}

// ---------------------------------------------------------------------------
extern "C" void kernel_launch(void* const* d_in, const int* in_sizes, int n_in,
                              void* d_out, int out_size, void* d_ws, size_t ws_size,
                              hipStream_t stream) {
    const float* X = (const float*)d_in[0];   // [65536,128]
    const float* W = (const float*)d_in[1];   // [64,128,32]
    const float* b = (const float*)d_in[2];   // [64,32]
    float* Out  = (float*)d_out;              // [65536,32]
    float* Wavg = (float*)d_ws;               // [128,32]
    float* bavg = Wavg + N_FEAT * N_CLS;      // [32]

    (void)in_sizes; (void)n_in; (void)out_size; (void)ws_size;

    forest_avg_wb<<<(N_FEAT * N_CLS + 255) / 256, 256, 0, stream>>>(W, b, Wavg, bavg);

    forest_gemm<<<BATCH / ROWS_PER_BLOCK, THREADS_PER_BLOCK, 0, stream>>>(
        X, Wavg, bavg, Out);
}